// graph_prompt_layer_mean_6468220748681
// MI455X (gfx1250) — compile-verified
//
#include <hip/hip_runtime.h>

typedef __attribute__((ext_vector_type(2))) float v2f;
typedef __attribute__((ext_vector_type(8))) float v8f;

#define GDIM 256  // node feature dim (matches reference DIM)

// ---------------------------------------------------------------------------
// Kernel 1: exclusive prefix sum of graph_len -> offs[], plus max(graph_len).
// Single block of 256 threads; each thread handles a contiguous chunk.
// ---------------------------------------------------------------------------
__global__ __launch_bounds__(256) void scan_lens_kernel(
    const int* __restrict__ len, int n,
    int* __restrict__ offs, int* __restrict__ maxOut) {
  __shared__ int part[256];
  __shared__ int smax[256];
  const int t = threadIdx.x;
  const int chunk = (n + 255) >> 8;
  const int base = t * chunk;

  int s = 0, m = 0;
  for (int i = 0; i < chunk; ++i) {
    const int idx = base + i;
    const int v = (idx < n) ? len[idx] : 0;
    s += v;
    m = v > m ? v : m;
  }
  part[t] = s;
  smax[t] = m;
  __syncthreads();

  if (t == 0) {
    int acc = 0, mm = 0;
    for (int i = 0; i < 256; ++i) {
      const int v = part[i];
      part[i] = acc;       // exclusive block offset
      acc += v;
      mm = smax[i] > mm ? smax[i] : mm;
    }
    maxOut[0] = mm;
  }
  __syncthreads();

  int off = part[t];
  for (int i = 0; i < chunk; ++i) {
    const int idx = base + i;
    if (idx < n) {
      offs[idx] = off;
      off += len[idx];
    }
  }
}

// ---------------------------------------------------------------------------
// Kernel 2: per-graph segment sum via V_WMMA_F32_16X16X4_F32, A = ones.
//   D(16x16) = ones(16x4) x B(4x16) + C  =>  every row of D = column sums.
// One block per graph; 16 waves x 16 columns = 256 dims.
//
// Main loop: 8 rows (two 4x16 tiles) per iteration with UNCONDITIONAL loads
// feeding two independent accumulator chains -> 4 loads in flight before any
// wait, no exec-mask branches in the hot path. Tail (<8 rows) is predicated.
// ---------------------------------------------------------------------------
__global__ __launch_bounds__(512) void seg_mean_wmma_kernel(
    const float* __restrict__ emb, const int* __restrict__ len,
    const int* __restrict__ offs, const int* __restrict__ maxPtr,
    float* __restrict__ out) {
  const int b    = blockIdx.x;
  const int tid  = threadIdx.x;
  const int wave = tid >> 5;       // 0..15
  const int lane = tid & 31;
  const int col0 = wave << 4;      // 16-column chunk per wave
  const int half = lane >> 4;      // 0: rows {r, r+1}; 1: rows {r+2, r+3}
  const int c    = lane & 15;

  const int   n     = len[b];
  const int   start = offs[b];
  const float maxf  = (float)maxPtr[0];

  const v2f onesA = {1.0f, 1.0f};
  v8f acc0 = {};
  v8f acc1 = {};

  // This lane's element pointer for tile starting at row 0.
  const float* base = emb + (size_t)start * GDIM + col0 + c
                      + (size_t)(half << 1) * GDIM;

  const int n_main = n & ~7;       // rows handled without predication

  const float* p = base;
  for (int r = 0; r < n_main; r += 8) {
    // Tile 0: rows r..r+3, Tile 1: rows r+4..r+7 — all in-bounds.
    v2f b0, b1;
    b0.x = p[0];
    b0.y = p[GDIM];
    b1.x = p[4 * GDIM];
    b1.y = p[5 * GDIM];
    acc0 = __builtin_amdgcn_wmma_f32_16x16x4_f32(
        false, onesA, false, b0, (short)0, acc0, false, false);
    acc1 = __builtin_amdgcn_wmma_f32_16x16x4_f32(
        false, onesA, false, b1, (short)0, acc1, false, false);
    p += 8 * GDIM;
  }

  // Tail: up to 7 rows, predicated (value-select; EXEC re-converged at WMMA).
  for (int r = n_main; r < n; r += 4) {
    const int row0 = r + (half << 1);
    const float* pt = base + (size_t)(r - (half << 1)) * GDIM
                      + (size_t)(half << 1) * GDIM;  // == row r tile, this lane
    v2f bt;
    bt.x = (row0     < n) ? pt[0]    : 0.0f;
    bt.y = (row0 + 1 < n) ? pt[GDIM] : 0.0f;
    acc0 = __builtin_amdgcn_wmma_f32_16x16x4_f32(
        false, onesA, false, bt, (short)0, acc0, false, false);
  }

  // Merge the two chains (8 VALU adds), then scale and store.
  v8f acc = acc0 + acc1;

  // D row M=0 lives in VGPR0, lanes 0..15 (N = lane). All D rows are equal,
  // so lanes 0..15 carry the 16 column sums of this wave's chunk.
  if (lane < 16) {
    out[(size_t)b * GDIM + col0 + lane] = acc[0] / maxf;  // match ref: sums / max
  }
}

// ---------------------------------------------------------------------------
// Host-side launcher. d_in[0] = graph_embedding (f32, total*256),
// d_in[1] = graph_len (i32, bsz). d_out = bsz*256 f32.
// d_ws: [0, bsz) ints = exclusive offsets, [bsz] = max length.
// ---------------------------------------------------------------------------
extern "C" void kernel_launch(void* const* d_in, const int* in_sizes, int n_in,
                              void* d_out, int out_size, void* d_ws, size_t ws_size,
                              hipStream_t stream) {
  const float* emb  = (const float*)d_in[0];
  const int*   glen = (const int*)d_in[1];
  const int    bsz  = in_sizes[1];

  int* offs   = (int*)d_ws;
  int* maxOut = offs + bsz;

  scan_lens_kernel<<<1, 256, 0, stream>>>(glen, bsz, offs, maxOut);
  seg_mean_wmma_kernel<<<bsz, 512, 0, stream>>>(emb, glen, offs, maxOut,
                                                (float*)d_out);
}